// NormalMixtureEM_65764539236703
// MI455X (gfx1250) — compile-verified
//
#include <hip/hip_runtime.h>
#include <hip/hip_bf16.h>
#include <math.h>

// NormalMixtureEM on MI455X (gfx1250).
// Key insight: the reference's EM loop never updates (mu, sigma, w), so all
// N_ITERS iterations produce identical (g, p, a, b) -> one pass is exact.
// Memory-bound: ~84 MB traffic (67 MB is the g output) => ~3.6us at 23.3 TB/s.
// Pass-1 row moments (sum x, sum x^2 over L=512) are computed with
// V_WMMA_F32_16X16X4_F32 using B = ones(4x16): every column of C accumulates
// exact fp32 row sums (16 rows per wave), which is both correct and exercises
// the CDNA5 matrix pipe. Streaming data (windows in, g out) uses non-temporal
// cache hints (TH=NT) so the 192MB L2 isn't polluted by write-once traffic.

typedef __attribute__((ext_vector_type(2))) float v2f;
typedef __attribute__((ext_vector_type(4))) float v4f;
typedef __attribute__((ext_vector_type(8))) float v8f;

#define EPSF 1e-8f
#define LL   512
#define KK   4
#define ROWS 16
#define TPAD 516  // LDS row pitch (floats): 516 % 64 == 4 -> WMMA A reads are bank-conflict-free

__device__ __forceinline__ float sigmoidf_(float x) { return 1.0f / (1.0f + __expf(-x)); }

__global__ __launch_bounds__(128)
void nmix_em_gfx1250_kernel(const float* __restrict__ windows,
                            const float* __restrict__ noise,
                            const float* __restrict__ centers,
                            const float* __restrict__ scales,
                            const float* __restrict__ initw,
                            const float* __restrict__ priorwp,
                            const float* __restrict__ logw,
                            const float* __restrict__ blendp,
                            float* __restrict__ out_g,
                            float* __restrict__ out_p,
                            float* __restrict__ out_a,
                            float* __restrict__ out_b)
{
    __shared__ float tile[ROWS][TPAD];          // 16 x 512 window tile (padded)
    __shared__ float wexp[LL];                  // exp(log_weights[l])
    __shared__ float smean[ROWS], sstd[ROWS], svar[ROWS];
    __shared__ float psum[4][ROWS], psq[4][ROWS];  // per-wave WMMA partial moments
    __shared__ float part[ROWS][4][12];         // per-row per-wave {S0[4],S1[4],S2[4]}

    const int tid     = threadIdx.x;
    const int lane    = tid & 31;
    const int wv      = tid >> 5;
    const int rowBase = blockIdx.x * ROWS;

    __builtin_prefetch(noise + (size_t)rowBase * KK, 0, 3);  // global_prefetch_b8

    // ---- stage tile into LDS (coalesced, non-temporal 16B loads) ----
    const float* wbase = windows + (size_t)rowBase * LL;
    for (int i = tid; i < ROWS * (LL / 4); i += 128) {
        int r = i >> 7;            // 128 float4 per row
        int c = i & 127;
        v4f v = __builtin_nontemporal_load(
            reinterpret_cast<const v4f*>(wbase + (size_t)r * LL) + c);
        *reinterpret_cast<v4f*>(&tile[r][c * 4]) = v;
    }
    for (int i = tid; i < LL; i += 128) wexp[i] = __expf(logw[i]);
    __syncthreads();

    // ---- pass 1: row sums & sums-of-squares via V_WMMA_F32_16X16X4_F32 ----
    // A: lane holds row (lane&15), K-pair (lane>>4)*2 of a 4-wide chunk.
    // B = ones(4x16) -> C[m][n] = rowsum(m) in every column n.
    {
        const int arow = lane & 15;
        const int kp   = (lane >> 4) << 1;
        v2f bones; bones.x = 1.0f; bones.y = 1.0f;
        v8f cs = {0.f, 0.f, 0.f, 0.f, 0.f, 0.f, 0.f, 0.f};
        v8f cq = {0.f, 0.f, 0.f, 0.f, 0.f, 0.f, 0.f, 0.f};
        for (int ch = wv * 32; ch < wv * 32 + 32; ++ch) {
            const float* ap = &tile[arow][ch * 4 + kp];
            v2f a;  a.x  = ap[0];        a.y  = ap[1];
            v2f a2; a2.x = a.x * a.x;    a2.y = a.y * a.y;
            cs = __builtin_amdgcn_wmma_f32_16x16x4_f32(false, a,  false, bones, (short)0, cs, false, false);
            cq = __builtin_amdgcn_wmma_f32_16x16x4_f32(false, a2, false, bones, (short)0, cq, false, false);
        }
        // lanes 0..15 hold rows 0..7 in c[0..7]; lanes 16..31 hold rows 8..15
        if ((lane & 15) == 0) {
            int rb = (lane >> 4) * 8;
            for (int v = 0; v < 8; ++v) { psum[wv][rb + v] = cs[v]; psq[wv][rb + v] = cq[v]; }
        }
    }
    __syncthreads();

    if (tid < ROWS) {
        float s = psum[0][tid] + psum[1][tid] + psum[2][tid] + psum[3][tid];
        float q = psq[0][tid]  + psq[1][tid]  + psq[2][tid]  + psq[3][tid];
        float mean = s * (1.0f / 512.0f);
        float var  = fmaxf((q - s * mean) * (1.0f / 511.0f), 0.0f);  // ddof=1
        smean[tid] = mean; svar[tid] = var; sstd[tid] = sqrtf(var);
    }
    __syncthreads();

    // ---- pass 2: responsibilities + weighted moment accumulation ----
    const float priorW = sigmoidf_(priorwp[0]);
    const float blend  = sigmoidf_(blendp[0]);
    float cent[KK], ascl[KK], iw[KK];
    for (int k = 0; k < KK; ++k) { cent[k] = centers[k]; ascl[k] = fabsf(scales[k]); iw[k] = initw[k]; }

    for (int r = 0; r < ROWS; ++r) {
        const int   row  = rowBase + r;
        const float mean = smean[r], stdv = sstd[r];
        float mu[KK], rs[KK], lw[KK];
        for (int k = 0; k < KK; ++k) {
            float nz = noise[(size_t)row * KK + k];
            mu[k] = cent[k] * stdv + mean + nz * stdv * 0.01f;
            rs[k] = 1.0f / (ascl[k] * stdv + EPSF);
            lw[k] = __logf((iw[k] + EPSF) * rs[k]);
        }
        float s0[KK] = {0, 0, 0, 0}, s1[KK] = {0, 0, 0, 0}, s2[KK] = {0, 0, 0, 0};
        float* gout = out_g + (size_t)row * (LL * KK);

        for (int it = 0; it < 4; ++it) {
            int   l  = wv * 128 + it * 32 + lane;   // wave-coalesced columns
            float x  = tile[r][l];
            float wl = wexp[l];
            float lm[KK]; float mmax = -3.4e38f;
            for (int k = 0; k < KK; ++k) {
                float z = (x - mu[k]) * rs[k];
                lm[k] = lw[k] - 0.5f * z * z;
                mmax = fmaxf(mmax, lm[k]);
            }
            float e[KK], se = 0.f;
            for (int k = 0; k < KK; ++k) { e[k] = __expf(lm[k] - mmax); se += e[k]; }
            float inv = 1.0f / se;
            v4f gv; gv.x = e[0] * inv; gv.y = e[1] * inv; gv.z = e[2] * inv; gv.w = e[3] * inv;
            // 512B-coalesced per wave, non-temporal (write-once streaming output)
            __builtin_nontemporal_store(gv, reinterpret_cast<v4f*>(gout + (size_t)l * KK));
            float gg[KK] = {gv.x, gv.y, gv.z, gv.w};
            for (int k = 0; k < KK; ++k) {
                float wg = gg[k] * wl;
                s0[k] += wg; s1[k] += wg * x; s2[k] += wg * x * x;
            }
        }
        // wave32 butterfly reduction of the 12 accumulators
        for (int off = 16; off >= 1; off >>= 1) {
            for (int k = 0; k < KK; ++k) {
                s0[k] += __shfl_xor(s0[k], off, 32);
                s1[k] += __shfl_xor(s1[k], off, 32);
                s2[k] += __shfl_xor(s2[k], off, 32);
            }
        }
        if (lane == 0) {
            for (int k = 0; k < KK; ++k) {
                part[r][wv][k]     = s0[k];
                part[r][wv][4 + k] = s1[k];
                part[r][wv][8 + k] = s2[k];
            }
        }
    }
    __syncthreads();

    // ---- finalize p, a, b (one thread per row) ----
    if (tid < ROWS) {
        const int   r = tid, row = rowBase + r;
        const float mean = smean[r], varr = svar[r];
        float pt[KK], psumk = 0.f, av[KK], bv[KK];
        for (int k = 0; k < KK; ++k) {
            float S0 = part[r][0][k]     + part[r][1][k]     + part[r][2][k]     + part[r][3][k];
            float S1 = part[r][0][4 + k] + part[r][1][4 + k] + part[r][2][4 + k] + part[r][3][4 + k];
            float S2 = part[r][0][8 + k] + part[r][1][8 + k] + part[r][2][8 + k] + part[r][3][8 + k];
            float sg = fmaxf(S0, EPSF);
            pt[k] = sg + priorW; psumk += pt[k];
            float dmu = S1 / sg;
            float a   = dmu * blend + mean * (1.0f - blend);
            float dvar = (S2 - 2.0f * a * S1 + a * a * S0) / sg;  // sum wg*(x-a)^2
            av[k] = a;
            bv[k] = sqrtf(dvar * blend + varr * (1.0f - blend) + EPSF);
        }
        float pinv = 1.0f / psumk;
        for (int k = 0; k < KK; ++k) {
            out_p[(size_t)row * KK + k] = pt[k] * pinv;
            out_a[(size_t)row * KK + k] = av[k];
            out_b[(size_t)row * KK + k] = bv[k];
        }
    }
}

extern "C" void kernel_launch(void* const* d_in, const int* in_sizes, int n_in,
                              void* d_out, int out_size, void* d_ws, size_t ws_size,
                              hipStream_t stream) {
    const float* windows = (const float*)d_in[0];
    const float* noise   = (const float*)d_in[1];
    const float* centers = (const float*)d_in[2];
    const float* scales  = (const float*)d_in[3];
    const float* initw   = (const float*)d_in[4];
    const float* priorwp = (const float*)d_in[5];
    const float* logw    = (const float*)d_in[6];
    const float* blendp  = (const float*)d_in[7];

    const size_t B = (size_t)in_sizes[0] / LL;   // 8192

    float* out   = (float*)d_out;
    float* out_g = out;                          // B*L*K
    float* out_p = out_g + B * LL * KK;          // B*K
    float* out_a = out_p + B * KK;               // B*K
    float* out_b = out_a + B * KK;               // B*K

    dim3 grid((unsigned)(B / ROWS));
    dim3 block(128);
    nmix_em_gfx1250_kernel<<<grid, block, 0, stream>>>(
        windows, noise, centers, scales, initw, priorwp, logw, blendp,
        out_g, out_p, out_a, out_b);
}